// KNNConnector_78340203479526
// MI455X (gfx1250) — compile-verified
//
#include <hip/hip_runtime.h>

// KNNConnector on gfx1250:
//   d(i,j) = |pi|^2 - 2 pi.pj + |pj|^2
// Augmented K=4 GEMM trick: A_j = (x,y,z, |pj|^2 or 1e30 if masked),
// B_i = (-2x,-2y,-2z, 1)  =>  dot = d(i,j) - |pi|^2  (same per-row ordering).
// One V_WMMA_F32_16X16X4_F32 per 16x16 tile; per-lane register top-16,
// lane-pair merge via LDS, int32 edge lists out.

#define N_NODES 8192
#define KSEL    16
#define TILES   (N_NODES / 16)

typedef __attribute__((ext_vector_type(2))) float v2f;
typedef __attribute__((ext_vector_type(8))) float v8f;

__global__ __launch_bounds__(256) void knn_wmma_kernel(
    const float* __restrict__ p,
    const unsigned char* __restrict__ mask,
    int* __restrict__ out)
{
  __shared__ float sval[8][32][KSEL];
  __shared__ int   sidx[8][32][KSEL];

  const int lane = threadIdx.x & 31;
  const int wave = threadIdx.x >> 5;
  const int half = lane >> 4;          // 0: K=0,1 / M=0..7   1: K=2,3 / M=8..15
  const int l16  = lane & 15;
  const int q    = (int)blockIdx.x * 128 + wave * 16 + l16;   // this lane's query

  // B operand (4x16): column N = l16, K components 2*half, 2*half+1
  const float qx = p[3 * q + 0];
  const float qy = p[3 * q + 1];
  const float qz = p[3 * q + 2];
  v2f b;
  b.x = half ? (-2.0f * qz) : (-2.0f * qx);
  b.y = half ? 1.0f         : (-2.0f * qy);

  // Per-lane sorted (ascending) top-16 candidate list in registers.
  float val[KSEL];
  int   idx[KSEL];
#pragma unroll
  for (int t = 0; t < KSEL; ++t) { val[t] = 3.0e38f; idx[t] = N_NODES - 1; }
  float worst = val[KSEL - 1];

  const int moff = half * 8;           // D rows this lane owns: moff..moff+7

  for (int jt = 0; jt < TILES; ++jt) {
    const int j0 = jt * 16;
    const int j  = j0 + l16;

    // A operand (16x4): row M = l16, K components 2*half, 2*half+1
    const float x = p[3 * j + 0];
    const float y = p[3 * j + 1];
    const float z = p[3 * j + 2];
    const float s = mask[j] ? (x * x + y * y + z * z) : 1.0e30f;
    v2f a;
    a.x = half ? z : x;
    a.y = half ? s : y;

    v8f c = {};
    // D[m][n] = A_j(m) . B_q(n) = d(q,j) - |pq|^2   (shifted distance)
    v8f dmat = __builtin_amdgcn_wmma_f32_16x16x4_f32(
        /*neg_a=*/false, a, /*neg_b=*/false, b,
        /*c_mod=*/(short)0, c, /*reuse_a=*/false, /*reuse_b=*/false);

#pragma unroll
    for (int v = 0; v < 8; ++v) {
      float cv = dmat[v];
      int   cj = j0 + moff + v;
      if (cv < worst) {
        // Bubble insert keeping ascending order; equal values keep the
        // earlier (lower-index) entry first, matching top_k tie-breaking.
#pragma unroll
        for (int t = 0; t < KSEL; ++t) {
          const bool  sm = cv < val[t];
          const float tv = val[t];
          const int   ti = idx[t];
          val[t] = sm ? cv : tv;
          idx[t] = sm ? cj : ti;
          cv = sm ? tv : cv;
          cj = sm ? ti : cj;
        }
        worst = val[KSEL - 1];
      }
    }
  }

  // Publish per-lane lists; lane pair (L, L+16) covers all 16 M's per tile.
#pragma unroll
  for (int t = 0; t < KSEL; ++t) {
    sval[wave][lane][t] = val[t];
    sidx[wave][lane][t] = idx[t];
  }
  __syncthreads();

  if (half == 0) {
    const float* av = sval[wave][lane];
    const int*   ai = sidx[wave][lane];
    const float* bv = sval[wave][lane + 16];
    const int*   bi = sidx[wave][lane + 16];

    const bool valid = mask[q] != 0;
    int* out_s = out;                       // senders   [N*K]
    int* out_r = out + N_NODES * KSEL;      // receivers [N*K]

    int pa = 0, pb = 0;
#pragma unroll
    for (int t = 0; t < KSEL; ++t) {
      const float va = av[pa];
      const float vb = bv[pb];
      const int   ia = ai[pa];
      const int   ib = bi[pb];
      const bool takeA = (va < vb) || (va == vb && ia < ib);
      const int  sel   = takeA ? ia : ib;
      pa += takeA ? 1 : 0;
      pb += takeA ? 0 : 1;
      out_s[q * KSEL + t] = valid ? sel : (N_NODES - 1);
      out_r[q * KSEL + t] = valid ? q   : (N_NODES - 1);
    }
  }
}

extern "C" void kernel_launch(void* const* d_in, const int* in_sizes, int n_in,
                              void* d_out, int out_size, void* d_ws, size_t ws_size,
                              hipStream_t stream) {
  (void)in_sizes; (void)n_in; (void)d_ws; (void)ws_size; (void)out_size;
  const float* p = (const float*)d_in[0];
  const unsigned char* mask = (const unsigned char*)d_in[1];  // JAX bool = 1 byte
  // d_in[2] is k (==16), baked into the kernel tiling.
  int* out = (int*)d_out;

  // 64 blocks x 256 threads (8 waves); each wave owns a 16-query tile and
  // streams all 512 neighbor tiles through v_wmma_f32_16x16x4_f32.
  knn_wmma_kernel<<<64, 256, 0, stream>>>(p, mask, out);
}